// ScaledDotProductAttention_1657857376380
// MI455X (gfx1250) — compile-verified
//
#include <hip/hip_runtime.h>
#include <math.h>

typedef float v2f __attribute__((ext_vector_type(2)));
typedef float v8f __attribute__((ext_vector_type(8)));

#define B_   8
#define H_   12
#define S_   1024
#define D_   64
#define BH_  (B_ * H_)
#define QT   16                 // query rows per block
#define KT   16                 // key columns per tile
#define NKT  (S_ / KT)          // 64 key tiles per row strip
#define NWAVES 4
#define TPB  (NWAVES * 32)      // 128 threads = 4 wave32
#define TILES_PER_WAVE (NKT / NWAVES)   // 16
#define LSTR 68                 // K/V tile LDS row stride (floats)
#define SSTR 1028               // strip row stride: %64==4 -> bank-conflict-free,
                                // *4B is 16B aligned for b128 access
#define SMEM_DYN (QT * SSTR * 4)  // 65,792 B dynamic LDS score strip

__device__ __forceinline__ float rmax16(float v) {
#pragma unroll
  for (int off = 1; off < 16; off <<= 1)
    v = fmaxf(v, __shfl_xor(v, off, 32));   // stays within each 16-lane half
  return v;
}
__device__ __forceinline__ float rsum16(float v) {
#pragma unroll
  for (int off = 1; off < 16; off <<= 1)
    v += __shfl_xor(v, off, 32);
  return v;
}

__global__ __launch_bounds__(TPB) void sdpa_kernel(
    const float* __restrict__ Q, const float* __restrict__ K,
    const float* __restrict__ V, const unsigned char* __restrict__ M,
    float* __restrict__ ctx_out, float* __restrict__ attn_out) {
  __shared__ float lq[QT][LSTR];               // Q tile staging
  __shared__ float lk[NWAVES][KT][LSTR];       // per-wave K (then V) tile
  __shared__ float lms[NWAVES][QT][2];         // per-wave row (max, sum)
  __shared__ float lrow[QT][2];                // final row (max, 1/sum)
  extern __shared__ float ls[];                // score strip [QT][SSTR] (dynamic)

  const int bh = blockIdx.x / (S_ / QT);
  const int qt = blockIdx.x % (S_ / QT);
  const int q0 = qt * QT;
  const int tid  = threadIdx.x;
  const int w    = tid >> 5;
  const int lane = tid & 31;
  const int g    = lane >> 4;    // lane half (0/1)
  const int n    = lane & 15;    // N (and A-row) index
  const int koff = 2 * g;

  const float* Qb = Q + (size_t)bh * S_ * D_;
  const float* Kb = K + (size_t)bh * S_ * D_;
  const float* Vb = V + (size_t)bh * S_ * D_;
  const unsigned char* Mb = M + (size_t)bh * S_ * S_;
  float* Cb = ctx_out + (size_t)bh * S_ * D_;
  float* Ab = attn_out + (size_t)bh * S_ * S_;

  // ---- stage Q tile (16x64 f32, coalesced float4) ----
#pragma unroll
  for (int i = 0; i < 2; ++i) {
    int idx = tid + i * TPB;          // 0..255 float4 slots
    int r = idx >> 4, c4 = idx & 15;
    *(float4*)(&lq[r][c4 * 4]) =
        *(const float4*)(Qb + (size_t)(q0 + r) * D_ + c4 * 4);
  }
  __syncthreads();

  // ---- hoist Q A-fragments into registers (invariant across key tiles) ----
  // A 16x4 f32 layout: lanes 0-15 hold M=lane K={0,1}; lanes 16-31 K={2,3}
  v2f qa[16];
#pragma unroll
  for (int s4 = 0; s4 < 16; ++s4) {
    qa[s4].x = lq[n][4 * s4 + koff];
    qa[s4].y = lq[n][4 * s4 + koff + 1];
  }

  // ======= sweep 1: scores -> strip, per-lane running max =======
  float vmax[8];
#pragma unroll
  for (int r = 0; r < 8; ++r) vmax[r] = -INFINITY;

  for (int t = 0; t < TILES_PER_WAVE; ++t) {
    const int k0 = (w + t * NWAVES) * KT;
    __builtin_amdgcn_wave_barrier();
#pragma unroll
    for (int i = 0; i < 8; ++i) {      // stage K tile
      int idx = lane + i * 32;
      int r = idx >> 4, c4 = idx & 15;
      *(float4*)(&lk[w][r][c4 * 4]) =
          *(const float4*)(Kb + (size_t)(k0 + r) * D_ + c4 * 4);
    }
    __builtin_amdgcn_wave_barrier();
    // 16x16 score tile: 16 chained V_WMMA_F32_16X16X4_F32 (K-dim 64)
    v8f c = {0.f, 0.f, 0.f, 0.f, 0.f, 0.f, 0.f, 0.f};
#pragma unroll
    for (int s4 = 0; s4 < 16; ++s4) {
      v2f b;   // B 4x16 (K^T): column N = lane%16, K split across lane halves
      b.x = lk[w][n][4 * s4 + koff];
      b.y = lk[w][n][4 * s4 + koff + 1];
      c = __builtin_amdgcn_wmma_f32_16x16x4_f32(false, qa[s4], false, b,
                                                (short)0, c, false, false);
    }
#pragma unroll
    for (int r = 0; r < 8; ++r) {
      const int qr = r + 8 * g;
      float sc = c[r] * 0.125f;   // 1/sqrt(64)
      if (Mb[(size_t)(q0 + qr) * S_ + k0 + n]) sc = -1e9f;
      ls[qr * SSTR + k0 + n] = sc;
      vmax[r] = fmaxf(vmax[r], sc);
    }
  }

  // one-time row-max reduction (lanes -> waves)
#pragma unroll
  for (int r = 0; r < 8; ++r) {
    float tm = rmax16(vmax[r]);
    if (n == 0) lms[w][r + 8 * g][0] = tm;
  }
  __syncthreads();
  if (tid < QT) {
    float mf = -INFINITY;
    for (int wv = 0; wv < NWAVES; ++wv) mf = fmaxf(mf, lms[wv][tid][0]);
    lrow[tid][0] = mf;
  }
  __syncthreads();

  // === sweep 2: exponentiate strip + per-lane sums + ctx = p @ V ===
  float mfr[8], vsum[8];
#pragma unroll
  for (int r = 0; r < 8; ++r) { mfr[r] = lrow[r + 8 * g][0]; vsum[r] = 0.f; }

  v8f ctx[4];
#pragma unroll
  for (int nc = 0; nc < 4; ++nc)
    ctx[nc] = (v8f){0.f, 0.f, 0.f, 0.f, 0.f, 0.f, 0.f, 0.f};

  for (int t = 0; t < TILES_PER_WAVE; ++t) {
    const int k0 = (w + t * NWAVES) * KT;
    __builtin_amdgcn_wave_barrier();
#pragma unroll
    for (int i = 0; i < 8; ++i) {      // stage V tile
      int idx = lane + i * 32;
      int r = idx >> 4, c4 = idx & 15;
      *(float4*)(&lk[w][r][c4 * 4]) =
          *(const float4*)(Vb + (size_t)(k0 + r) * D_ + c4 * 4);
    }
    // exponentiate this tile's strip columns in place, accumulate sums
#pragma unroll
    for (int r = 0; r < 8; ++r) {
      const int qr = r + 8 * g;
      float p = __expf(ls[qr * SSTR + k0 + n] - mfr[r]);
      ls[qr * SSTR + k0 + n] = p;
      vsum[r] += p;
    }
    __builtin_amdgcn_wave_barrier();
    // ctx[16x64] += p_tile[16x16] @ V_tile[16x64]; K-dim 16 = 4 x (K=4)
#pragma unroll
    for (int nc = 0; nc < 4; ++nc) {
#pragma unroll
      for (int ss = 0; ss < 4; ++ss) {
        v2f a, b;
        a.x = ls[n * SSTR + k0 + 4 * ss + koff];       // A row m=n from strip
        a.y = ls[n * SSTR + k0 + 4 * ss + koff + 1];
        b.x = lk[w][4 * ss + koff][nc * 16 + n];
        b.y = lk[w][4 * ss + koff + 1][nc * 16 + n];
        ctx[nc] = __builtin_amdgcn_wmma_f32_16x16x4_f32(
            false, a, false, b, (short)0, ctx[nc], false, false);
      }
    }
  }

  // one-time row-sum reduction (lanes -> waves)
#pragma unroll
  for (int r = 0; r < 8; ++r) {
    float tsum = rsum16(vsum[r]);
    if (n == 0) lms[w][r + 8 * g][1] = tsum;
  }
  __syncthreads();
  if (tid < QT) {
    float sf = 0.f;
    for (int wv = 0; wv < NWAVES; ++wv) sf += lms[wv][tid][1];
    lrow[tid][1] = 1.f / sf;
  }
  __syncthreads();

  // === sweep 3: coalesced normalized attn writeback (b128 stores) ===
#pragma unroll
  for (int i = 0; i < QT * (S_ / 4) / TPB; ++i) {   // 32 float4 per thread
    int e = tid + i * TPB;            // float4 slot; 256 per row
    int r = e >> 8, c4 = e & 255;
    float inv = lrow[r][1];
    float4 p = *(const float4*)(&ls[r * SSTR + c4 * 4]);
    p.x *= inv; p.y *= inv; p.z *= inv; p.w *= inv;
    *(float4*)(Ab + (size_t)(q0 + r) * S_ + c4 * 4) = p;
  }

  // ---- row-normalize partial ctx, reduce across waves via LDS ----
  float invr[8];
#pragma unroll
  for (int r = 0; r < 8; ++r) invr[r] = lrow[r + 8 * g][1];
  __syncthreads();   // strip/lk reuse safety before overwriting lk
#pragma unroll
  for (int nc = 0; nc < 4; ++nc)
#pragma unroll
    for (int r = 0; r < 8; ++r)
      lk[w][r + 8 * g][nc * 16 + n] = ctx[nc][r] * invr[r];
  __syncthreads();
#pragma unroll
  for (int i = 0; i < 8; ++i) {
    int e = tid * 8 + i;                 // 0..1023 elements of 16x64 tile
    int r = e >> 6, c = e & 63;
    float s = lk[0][r][c] + lk[1][r][c] + lk[2][r][c] + lk[3][r][c];
    Cb[(size_t)(q0 + r) * D_ + c] = s;
  }
}

extern "C" void kernel_launch(void* const* d_in, const int* in_sizes, int n_in,
                              void* d_out, int out_size, void* d_ws,
                              size_t ws_size, hipStream_t stream) {
  (void)in_sizes; (void)n_in; (void)out_size; (void)d_ws; (void)ws_size;
  const float* Q = (const float*)d_in[0];
  const float* K = (const float*)d_in[1];
  const float* V = (const float*)d_in[2];
  const unsigned char* M = (const unsigned char*)d_in[3];  // bool mask
  float* ctx  = (float*)d_out;                                   // [B,H,S,D]
  float* attn = (float*)d_out + (size_t)B_ * H_ * S_ * D_;       // [B,H,S,S]
  dim3 grid(BH_ * (S_ / QT));   // 96 * 64 = 6144 blocks
  dim3 block(TPB);
  // 64.3 KB dynamic LDS score strip; CDNA5 WGP has 320 KB LDS.
  hipLaunchKernelGGL(sdpa_kernel, grid, block, SMEM_DYN, stream, Q, K, V, M,
                     ctx, attn);
}